// Net_26809185861951
// MI455X (gfx1250) — compile-verified
//
#include <hip/hip_runtime.h>

// Reference: out[b,l,:] = prototypes[labels[b,l], :]
//   B=512, L=21, NCHAN=1, T=29, F=129, N_PROTO=25
// Pure gather/copy. HBM cost = 160.9 MB of stores (~6.9 us at 23.3 TB/s);
// the 374 KB prototype table lives in L2/WGP$. Bandwidth-bound -> maximize
// per-wave MLP: immediate-offset load clause, single wait, store clause.

#define NUM_SLOTS   (512 * 21)     // B * L = 10752 gather slots
#define PROTO_ELEMS (29 * 129)     // NCHAN*T*F = 3741 floats per slot
#define THREADS     256            // 8 wave32 waves per block
#define FULL_ITERS  14             // 3741 = 14*256 + 157
#define TAIL_COUNT  (PROTO_ELEMS - FULL_ITERS * THREADS)   // 157

__global__ __launch_bounds__(THREADS) void proto_gather_kernel(
    const float* __restrict__ prototypes,   // (25, 3741)
    const int*   __restrict__ labels,       // (B*L,)  int32
    float*       __restrict__ out)          // (B*L, 3741)
{
    const unsigned slot = blockIdx.x;             // one (b,l) pair per block
    const unsigned tid  = threadIdx.x;

    // Block-uniform index -> scalar s_load (KMcnt), scale_offset addressing.
    const int lab = labels[slot];

    // Fixed per-lane bases; never mutated, so every access below is a pure
    // immediate-offset op (no address-VGPR WAR hazards -> no s_wait_xcnt).
    const float* __restrict__ src = prototypes + (size_t)lab  * PROTO_ELEMS + tid;
    float*       __restrict__ dst = out        + (size_t)slot * PROTO_ELEMS + tid;

    // ---- load clause: 14 independent b32 loads at immediate offsets ----
    float v[FULL_ITERS];
    #pragma unroll
    for (int k = 0; k < FULL_ITERS; ++k) {
        v[k] = src[k * THREADS];
    }

    // Predicated tail element (only lanes with tid < 157).
    const bool has_tail = tid < TAIL_COUNT;
    float vt = 0.0f;
    if (has_tail) {
        vt = src[FULL_ITERS * THREADS];
    }

    // ---- store clause: NT hint keeps the write-once stream out of L2's
    // hot set (prototype table stays resident). 32 lanes x b32 = full line.
    #pragma unroll
    for (int k = 0; k < FULL_ITERS; ++k) {
        __builtin_nontemporal_store(v[k], dst + k * THREADS);
    }
    if (has_tail) {
        __builtin_nontemporal_store(vt, dst + FULL_ITERS * THREADS);
    }
}

extern "C" void kernel_launch(void* const* d_in, const int* in_sizes, int n_in,
                              void* d_out, int out_size, void* d_ws, size_t ws_size,
                              hipStream_t stream) {
    // setup_inputs() order: inputs (unused by the reference math), labels, prototypes
    const float* inputs     = (const float*)d_in[0];
    const int*   labels     = (const int*)  d_in[1];
    const float* prototypes = (const float*)d_in[2];
    float*       out        = (float*)d_out;
    (void)inputs; (void)in_sizes; (void)n_in; (void)out_size; (void)d_ws; (void)ws_size;

    proto_gather_kernel<<<dim3(NUM_SLOTS), dim3(THREADS), 0, stream>>>(
        prototypes, labels, out);
}